// Base_Conv3D_76270029242785
// MI455X (gfx1250) — compile-verified
//
#include <hip/hip_runtime.h>

// CDNA5 / gfx1250, wave32. Full-precision modulated conv3d via V_WMMA_F32_16X16X4_F32.
// Block = one (b, t) and 2 consecutive h rows; 4 waves = 2 rows x 2 w-segments(64).
// Each wave holds 4 accumulator tiles (16cout x 16w each) -> 432 WMMAs/wave.
// Tap loop fully unrolled: fixed accumulator VGPRs, immediate LDS offsets.

typedef __attribute__((ext_vector_type(2))) float v2f;
typedef __attribute__((ext_vector_type(8))) float v8f;

#define B_    4
#define CIN_  16
#define COUT_ 16
#define T_    16
#define H_    128
#define W_    128
#define SCALE_ 0.04811252243246881f   // 1/sqrt(16*3*3*3)

#define XROW 144                       // padded w-length (float2 units); 144*2 % 64 == 32 -> bank-clean
#define NA   (27 * 4 * 32 * 2)         // 6912 floats: A fragments (tap, k-chunk, lane, 2)
#define NDH  4                         // dh halo extent for 2 output rows: h-1 .. h+2
#define NX   (3 * NDH * 8 * XROW * 2)  // 27648 floats: X tile [dt][dh][ciPair][w][2]

__global__ __launch_bounds__(128) void conv3d_mod_wmma_f32(
    const float* __restrict__ input,    // [B][CIN][T][H][W]
    const float* __restrict__ cond,     // [B][1][CIN][1][1][1]
    const float* __restrict__ weight,   // [COUT][CIN][3][3][3]
    float* __restrict__ out)            // [B][COUT][T][H][W]
{
    extern __shared__ float smem[];
    float* sA = smem;        // A fragments, ISA layout
    float* sX = smem + NA;   // halo tile

    const int bi  = blockIdx.x;
    const int h0  = (bi % (H_ / 2)) * 2;          // base output row (h0, h0+1)
    const int t   = (bi / (H_ / 2)) % T_;
    const int b   = bi / ((H_ / 2) * T_);
    const int tid = threadIdx.x;

    // --- Stage modulated tap matrices as ready-to-load A fragments ---------
    // sA[((tap*4 + c)*32 + lane)*2 + j]:
    //   lane<16 : (W[m][4c+0], W[m][4c+1]) ; lane>=16 : (W[m][4c+2], W[m][4c+3]), m = lane&15
    for (int j = tid; j < NA; j += 128) {
        int tap  = j >> 8;             // /256
        int rem  = j & 255;
        int c    = rem >> 6;           // k-chunk 0..3
        int rem2 = rem & 63;
        int l    = rem2 >> 1;          // lane 0..31
        int jj   = rem2 & 1;
        int cin  = 4 * c + 2 * (l >> 4) + jj;
        int m    = l & 15;
        int kt = tap / 9, kh = (tap / 3) % 3, kw = tap % 3;
        float wgt = weight[(((m * CIN_ + cin) * 3 + kt) * 3 + kh) * 3 + kw];
        sA[j] = wgt * SCALE_ * cond[b * CIN_ + cin];
    }

    // --- Stage input halo tile (zero-padded), ci pair-interleaved ----------
    // rows: cin(16) x dt(3) x dh(4); 130 meaningful w entries each
    for (int i = tid; i < 16 * 3 * NDH * 130; i += 128) {
        int r   = i / 130;
        int wi  = i - r * 130;         // 0..129 -> input w = wi-1
        int cin = r / (3 * NDH);
        int rr  = r - cin * (3 * NDH);
        int dt  = rr / NDH;
        int dh  = rr - dt * NDH;
        int ts = t + dt - 1, hs = h0 + dh - 1, ws = wi - 1;
        float v = 0.0f;
        if (((unsigned)ts < (unsigned)T_) & ((unsigned)hs < (unsigned)H_) &
            ((unsigned)ws < (unsigned)W_))
            v = input[((size_t)(b * CIN_ + cin) * T_ + ts) * (H_ * W_) +
                      (size_t)hs * W_ + ws];
        sX[(((dt * NDH + dh) * 8 + (cin >> 1)) * XROW + wi) * 2 + (cin & 1)] = v;
    }
    __syncthreads();

    // --- WMMA main loop: D[16co x 16n] += A_tap[16x4] * B[4x16] ------------
    const int wvid = tid >> 5;
    const int hr   = wvid >> 1;        // which of the 2 output rows
    const int wseg = (wvid & 1) * 64;  // 64-wide w segment
    const int lane = tid & 31;
    const int hf   = lane >> 4;        // half-wave
    const int lx   = lane & 15;

    v8f acc0 = {}, acc1 = {}, acc2 = {}, acc3 = {};

#pragma unroll
    for (int tap = 0; tap < 27; ++tap) {
        const int kt = tap / 9, kh = (tap / 3) % 3, kw = tap % 3;
        const int rowb = (kt * NDH + (kh + hr)) * 8;   // dh = kh + hr
        const int n0   = wseg + lx + kw;               // lds w-index = n + kw
#pragma unroll
        for (int c = 0; c < 4; ++c) {
            v2f a = *(const v2f*)&sA[(((tap * 4 + c) << 5) + lane) * 2];
            int base = (rowb + 2 * c + hf) * (XROW * 2);
            v2f b0 = *(const v2f*)&sX[base + n0 * 2];
            v2f b1 = *(const v2f*)&sX[base + (n0 + 16) * 2];
            v2f b2 = *(const v2f*)&sX[base + (n0 + 32) * 2];
            v2f b3 = *(const v2f*)&sX[base + (n0 + 48) * 2];
            acc0 = __builtin_amdgcn_wmma_f32_16x16x4_f32(
                false, a, false, b0, (short)0, acc0, false, false);
            acc1 = __builtin_amdgcn_wmma_f32_16x16x4_f32(
                false, a, false, b1, (short)0, acc1, false, false);
            acc2 = __builtin_amdgcn_wmma_f32_16x16x4_f32(
                false, a, false, b2, (short)0, acc2, false, false);
            acc3 = __builtin_amdgcn_wmma_f32_16x16x4_f32(
                false, a, false, b3, (short)0, acc3, false, false);
        }
    }

    // --- Store: VGPR r -> (M = r + 8*hf, N = lx), coalesced over w ---------
    const size_t strideC = (size_t)T_ * H_ * W_;
    float* o = out + (size_t)b * COUT_ * strideC + (size_t)t * (H_ * W_) +
               (size_t)(h0 + hr) * W_ + wseg + lx;
#pragma unroll
    for (int r = 0; r < 8; ++r) {
        size_t mo = (size_t)(hf * 8 + r) * strideC;
        o[mo]      = acc0[r];
        o[mo + 16] = acc1[r];
        o[mo + 32] = acc2[r];
        o[mo + 48] = acc3[r];
    }
}

extern "C" void kernel_launch(void* const* d_in, const int* in_sizes, int n_in,
                              void* d_out, int out_size, void* d_ws, size_t ws_size,
                              hipStream_t stream) {
    const float* input  = (const float*)d_in[0];
    const float* cond   = (const float*)d_in[1];
    const float* weight = (const float*)d_in[2];
    float* out = (float*)d_out;

    const int    nBlocks = B_ * T_ * (H_ / 2);    // 4096: one (b,t) x 2 h-rows each
    const size_t shmem   = (size_t)(NA + NX) * sizeof(float);  // 138,240 B

    conv3d_mod_wmma_f32<<<dim3(nBlocks), dim3(128), shmem, stream>>>(
        input, cond, weight, out);
}